// SingleScaleImageWarping_30227979829794
// MI455X (gfx1250) — compile-verified
//
#include <hip/hip_runtime.h>

#define S_ 4
#define B_ 8
#define C_ 3
#define H_ 384
#define W_ 384
#define EPS_ 1e-7f

typedef __attribute__((ext_vector_type(2))) float v2f;
typedef __attribute__((ext_vector_type(8))) float v8f;

// ---------------------------------------------------------------------------
// Setup kernel: one wave32. Computes per-batch Q (3x3) and t (3) such that
//   proj = depth * (Q @ [x,y,1]) + t
// Matrix products K@pose and P3@invK3 are done on the matrix pipe with
// V_WMMA_F32_16X16X4_F32, packing 4 batches per WMMA as block-diagonal tiles.
// ---------------------------------------------------------------------------
__global__ __launch_bounds__(32)
void setup_coeffs(const float* __restrict__ K0,    // intrinsics[0]: (B,4,4)
                  const float* __restrict__ poses, // (B,4,4)
                  float* __restrict__ coeff)       // (B,12): Q row-major, then t
{
    __shared__ float Ks [B_][4][4];
    __shared__ float Po [B_][4][4];
    __shared__ float Inv[B_][4][4];
    __shared__ float Pm [B_][4][4];   // P = K @ pose (rows 0..2 used)

    const int lane = threadIdx.x;
    for (int i = lane; i < B_ * 16; i += 32) {
        ((float*)Ks)[i] = K0[i];
        ((float*)Po)[i] = poses[i];
    }
    __syncthreads();

    // Per-lane 4x4 Gauss-Jordan inverse (lanes 0..7; diag of K is fx,fy,1,1)
    if (lane < B_) {
        float a[4][8];
        for (int i = 0; i < 4; i++)
            for (int j = 0; j < 4; j++) {
                a[i][j]     = Ks[lane][i][j];
                a[i][4 + j] = (i == j) ? 1.0f : 0.0f;
            }
        for (int col = 0; col < 4; col++) {
            float pinv = 1.0f / a[col][col];
            for (int j = 0; j < 8; j++) a[col][j] *= pinv;
            for (int row = 0; row < 4; row++)
                if (row != col) {
                    float f = a[row][col];
                    for (int j = 0; j < 8; j++) a[row][j] -= f * a[col][j];
                }
        }
        for (int i = 0; i < 4; i++)
            for (int j = 0; j < 4; j++) Inv[lane][i][j] = a[i][4 + j];
    }
    __syncthreads();

    // Lane decomposition per WMMA layout (A 16x4 f32: v[0]=K{0|2}, v[1]=K{1|3})
    const int m  = lane & 15;   // row index of A / col index of B
    const int kh = lane >> 4;   // half-wave selects K pair {0,1} vs {2,3}
    const int g  = m >> 2;      // 4x4 block (= batch within group of 4)
    const int i  = m & 3;       // row within block
    const int j  = m & 3;       // col within block (n == m for this lane)

    // Stage 1: P_g = K_g @ pose_g, 4 batches per WMMA
    for (int w = 0; w < 2; ++w) {
        const int bb = w * 4 + g;
        v2f A, Bv;
        v8f Cz = {};
        A[0]  = Ks[bb][i][2 * kh];
        A[1]  = Ks[bb][i][2 * kh + 1];
        Bv[0] = Po[bb][2 * kh][j];
        Bv[1] = Po[bb][2 * kh + 1][j];
        v8f D = __builtin_amdgcn_wmma_f32_16x16x4_f32(
            false, A, false, Bv, (short)0, Cz, false, false);
        for (int r = 0; r < 8; r++) {
            const int md = r + 8 * kh;
            if ((md >> 2) == g) Pm[w * 4 + g][md & 3][j] = D[r];
        }
    }
    __syncthreads();

    if (lane < B_) {  // t = P[:,3]
        coeff[lane * 12 + 9]  = Pm[lane][0][3];
        coeff[lane * 12 + 10] = Pm[lane][1][3];
        coeff[lane * 12 + 11] = Pm[lane][2][3];
    }

    // Stage 2: Q_g = P_g[:, :3] @ Inv_g[:3,:3]  (K padded to 4 with zeros)
    for (int w = 0; w < 2; ++w) {
        const int bb = w * 4 + g;
        const int k0 = 2 * kh, k1 = 2 * kh + 1;
        v2f A, Bv;
        v8f Cz = {};
        A[0]  = (k0 < 3) ? Pm[bb][i][k0] : 0.0f;
        A[1]  = (k1 < 3) ? Pm[bb][i][k1] : 0.0f;
        Bv[0] = (k0 < 3 && j < 3) ? Inv[bb][k0][j] : 0.0f;
        Bv[1] = (k1 < 3 && j < 3) ? Inv[bb][k1][j] : 0.0f;
        v8f D = __builtin_amdgcn_wmma_f32_16x16x4_f32(
            false, A, false, Bv, (short)0, Cz, false, false);
        for (int r = 0; r < 8; r++) {
            const int md = r + 8 * kh;
            const int im = md & 3;
            if ((md >> 2) == g && im < 3 && j < 3)
                coeff[(w * 4 + g) * 12 + im * 3 + j] = D[r];
        }
    }
}

// ---------------------------------------------------------------------------
// Main kernel: one thread per (s,b,y,x), 3 channels per thread.
// NT load for depth (no reuse), NT store for output (streaming); image0
// gathers use default RT hints so the 14 MB image stays hot in L1/L2.
// ---------------------------------------------------------------------------
__global__ __launch_bounds__(256)
void warp_kernel(const float* __restrict__ image0,  // (B,C,H,W)
                 const float* __restrict__ depths,  // (S,B,1,H,W)
                 const float* __restrict__ coeff,   // (B,12)
                 float* __restrict__ out)           // (S,B,C,H,W)
{
    const int sb = blockIdx.y;          // s*B + b  (b uniform per block)
    const int b  = sb & (B_ - 1);
    const int p  = blockIdx.x * 256 + threadIdx.x;  // pixel index in [0, H*W)
    const int x  = p % W_;
    const int y  = p / W_;

    const float* cf = coeff + b * 12;
    const float q00 = cf[0], q01 = cf[1], q02 = cf[2];
    const float q10 = cf[3], q11 = cf[4], q12 = cf[5];
    const float q20 = cf[6], q21 = cf[7], q22 = cf[8];
    const float t0  = cf[9], t1  = cf[10], t2 = cf[11];

    const float d = __builtin_nontemporal_load(
        depths + (size_t)sb * (H_ * W_) + p);

    const float fx = (float)x, fy = (float)y;
    const float r0 = fmaf(q00, fx, fmaf(q01, fy, q02));
    const float r1 = fmaf(q10, fx, fmaf(q11, fy, q12));
    const float r2 = fmaf(q20, fx, fmaf(q21, fy, q22));
    const float p0 = fmaf(d, r0, t0);
    const float p1 = fmaf(d, r1, t1);
    const float p2 = fmaf(d, r2, t2);

    const float inv = 1.0f / (p2 + EPS_);
    const float px  = p0 * inv;
    const float py  = p1 * inv;
    // literal replication of reference normalize->denormalize chain
    const float gx = (px / (float)(W_ - 1) - 0.5f) * 2.0f;
    const float gy = (py / (float)(H_ - 1) - 0.5f) * 2.0f;
    float ix = (gx + 1.0f) * (W_ * 0.5f) - 0.5f;
    float iy = (gy + 1.0f) * (H_ * 0.5f) - 0.5f;
    ix = fminf(fmaxf(ix, 0.0f), (float)(W_ - 1));
    iy = fminf(fmaxf(iy, 0.0f), (float)(H_ - 1));

    const float x0f = floorf(ix), y0f = floorf(iy);
    const float wx = ix - x0f, wy = iy - y0f;
    const int x0 = (int)x0f, y0 = (int)y0f;
    const int x1 = min(x0 + 1, W_ - 1), y1 = min(y0 + 1, H_ - 1);

    const float w00 = (1.0f - wx) * (1.0f - wy);
    const float w01 = wx * (1.0f - wy);
    const float w10 = (1.0f - wx) * wy;
    const float w11 = wx * wy;

    const float* img = image0 + (size_t)b * (C_ * H_ * W_);
    const size_t o00 = (size_t)y0 * W_ + x0;
    const size_t o01 = (size_t)y0 * W_ + x1;
    const size_t o10 = (size_t)y1 * W_ + x0;
    const size_t o11 = (size_t)y1 * W_ + x1;

    float* o = out + (size_t)sb * (C_ * H_ * W_) + p;
#pragma unroll
    for (int c = 0; c < C_; c++) {
        const float* ic = img + (size_t)c * (H_ * W_);
        const float v = ic[o00] * w00 + ic[o01] * w01 +
                        ic[o10] * w10 + ic[o11] * w11;
        __builtin_nontemporal_store(v, o + (size_t)c * (H_ * W_));
    }
}

extern "C" void kernel_launch(void* const* d_in, const int* in_sizes, int n_in,
                              void* d_out, int out_size, void* d_ws, size_t ws_size,
                              hipStream_t stream) {
    (void)in_sizes; (void)n_in; (void)out_size; (void)ws_size;
    const float* images     = (const float*)d_in[0];  // (S,B,C,H,W); image0 = first slice
    const float* depths     = (const float*)d_in[1];  // (S,B,1,H,W)
    const float* poses      = (const float*)d_in[2];  // (B,4,4)
    const float* intrinsics = (const float*)d_in[3];  // (S,B,4,4); K0 = first slice
    float* out   = (float*)d_out;
    float* coeff = (float*)d_ws;                      // B*12 floats

    setup_coeffs<<<1, 32, 0, stream>>>(intrinsics, poses, coeff);

    const int pix_blocks = (H_ * W_) / 256;           // 576
    dim3 grid(pix_blocks, S_ * B_);
    warp_kernel<<<grid, 256, 0, stream>>>(images, depths, coeff, out);
}